// GetModel_59176059404747
// MI455X (gfx1250) — compile-verified
//
#include <hip/hip_runtime.h>
#include <math.h>

#define BN_SCALE_F 0.9999950000374997f

typedef float v2f __attribute__((ext_vector_type(2)));
typedef float v8f __attribute__((ext_vector_type(8)));

// ---------------------------------------------------------------------------
// Dual-M WMMA tile: computes D0 = X[m0:m0+16] * Wp^T, D1 = X[m0+16:m0+32] * Wp^T
// sharing one B-fragment load per K step. Branchless; all 32 lanes active.
// A layout: lane = h*16 + m; VGPRs = X[m][k0+2h], X[m][k0+2h+1]
// B layout: lane = h*16 + n; VGPRs = Wp[n][k0+2h], Wp[n][k0+2h+1]  (B = W^T)
// C/D layout: lane = h*16 + n; acc[v] = D[v + 8h][n]
// ---------------------------------------------------------------------------
__device__ __forceinline__ void wmma_tile2_f32(
    const float* __restrict__ X, int ldx, int m0,
    const float* __restrict__ Wp, int kPad, int n0, int lane,
    v8f& acc0, v8f& acc1)
{
  const int h = lane >> 4;
  const int r = lane & 15;
  const float* xrow0 = X + (size_t)(m0 + r) * ldx + 2 * h;
  const float* xrow1 = xrow0 + (size_t)16 * ldx;
  const float* wrow  = Wp + (size_t)(n0 + r) * kPad + 2 * h;
  acc0 = {};
  acc1 = {};
#pragma unroll 4
  for (int k0 = 0; k0 < kPad; k0 += 4) {
    const v2f b  = *(const v2f*)(wrow + k0);
    const v2f a0 = *(const v2f*)(xrow0 + k0);
    const v2f a1 = *(const v2f*)(xrow1 + k0);
    acc0 = __builtin_amdgcn_wmma_f32_16x16x4_f32(
        false, a0, false, b, (short)0, acc0, false, false);
    acc1 = __builtin_amdgcn_wmma_f32_16x16x4_f32(
        false, a1, false, b, (short)0, acc1, false, false);
  }
}

__device__ __forceinline__ void wmma_epilogue(
    v8f acc, float* __restrict__ Y, int ldy, int m0, int n0,
    const float* __restrict__ bias, const float* __restrict__ gamma,
    const float* __restrict__ beta, int Cout, int relu, int lane)
{
  const int h = lane >> 4;
  const int r = lane & 15;
  const int n = n0 + r;
  if (n >= Cout) return;       // only fc3 (Cout=40) ever diverges here
  const float bb = bias[n];
  const bool hasbn = (gamma != nullptr);
  float g = 1.f, be = 0.f;
  if (hasbn) { g = gamma[n]; be = beta[n]; }
#pragma unroll
  for (int v = 0; v < 8; ++v) {
    const int m = m0 + v + 8 * h;
    float val = acc[v] + bb;
    if (hasbn) val = val * (BN_SCALE_F * g) + be;
    if (relu)  val = fmaxf(val, 0.f);
    Y[(size_t)m * ldy + n] = val;
  }
}

// ---------------------------------------------------------------------------
// Wp[n*kPad + k] = (n<Cout && k<cin) ? W[n*cin + k] : 0
// ---------------------------------------------------------------------------
__global__ void k_pad_weights(const float* __restrict__ W, int cin, int Cout,
                              float* __restrict__ Wp, int kPad, int CoutPad)
{
  int i = blockIdx.x * blockDim.x + threadIdx.x;
  int total = CoutPad * kPad;
  if (i >= total) return;
  int k = i % kPad;
  int n = i / kPad;
  Wp[i] = (n < Cout && k < cin) ? W[(size_t)n * cin + k] : 0.f;
}

// ---------------------------------------------------------------------------
// points [B,6,N] -> xyz [B,N,3], nrm [B,N,3]
// ---------------------------------------------------------------------------
__global__ void k_split_points(const float* __restrict__ pts,
                               float* __restrict__ xyz, float* __restrict__ nrm,
                               int B, int N)
{
  int i = blockIdx.x * blockDim.x + threadIdx.x;
  int total = B * N * 3;
  if (i >= total) return;
  int c = i % 3;
  int n = (i / 3) % N;
  int b = i / (3 * N);
  xyz[i] = pts[(size_t)b * 6 * N + (size_t)c * N + n];
  nrm[i] = pts[(size_t)b * 6 * N + (size_t)(c + 3) * N + n];
}

// ---------------------------------------------------------------------------
// Farthest point sampling; one block per batch. First-index tie-break (argmax).
// ---------------------------------------------------------------------------
__global__ void k_fps(const float* __restrict__ xyz, int N, int S, int* __restrict__ out)
{
  __shared__ float dist[2048];
  __shared__ float rv[256];
  __shared__ int   ri[256];
  __shared__ int   s_far;
  const int b = blockIdx.x;
  const int tid = threadIdx.x;
  const float* x = xyz + (size_t)b * N * 3;
  for (int n = tid; n < N; n += blockDim.x) dist[n] = 1e10f;
  if (tid == 0) s_far = 0;
  __syncthreads();
  for (int it = 0; it < S; ++it) {
    const int far = s_far;
    if (tid == 0) out[b * S + it] = far;
    const float cx = x[far * 3 + 0];
    const float cy = x[far * 3 + 1];
    const float cz = x[far * 3 + 2];
    for (int n = tid; n < N; n += blockDim.x) {
      const float dx = x[n * 3 + 0] - cx;
      const float dy = x[n * 3 + 1] - cy;
      const float dz = x[n * 3 + 2] - cz;
      const float d = dx * dx + dy * dy + dz * dz;
      dist[n] = fminf(dist[n], d);
    }
    __syncthreads();
    float bv = -1.f;
    int   bi = 0;
    for (int n = tid; n < N; n += blockDim.x) {
      const float d = dist[n];
      if (d > bv) { bv = d; bi = n; }
    }
    rv[tid] = bv;
    ri[tid] = bi;
    __syncthreads();
    for (int s2 = 128; s2 > 0; s2 >>= 1) {
      if (tid < s2) {
        if (rv[tid + s2] > rv[tid] ||
            (rv[tid + s2] == rv[tid] && ri[tid + s2] < ri[tid])) {
          rv[tid] = rv[tid + s2];
          ri[tid] = ri[tid + s2];
        }
      }
      __syncthreads();
    }
    if (tid == 0) s_far = ri[0];
    __syncthreads();
  }
}

// ---------------------------------------------------------------------------
// new_xyz[b,s,:] = xyz[b, fps_idx[b,s], :]
// ---------------------------------------------------------------------------
__global__ void k_gather_xyz(const float* __restrict__ xyz, const int* __restrict__ idx,
                             int N, int S, int B, float* __restrict__ out)
{
  int i = blockIdx.x * blockDim.x + threadIdx.x;
  int total = B * S * 3;
  if (i >= total) return;
  int c = i % 3;
  int s = (i / 3) % S;
  int b = i / (3 * S);
  out[i] = xyz[((size_t)b * N + idx[b * S + s]) * 3 + c];
}

// ---------------------------------------------------------------------------
// Ball query: first nsample in-radius indices ascending, padded with first.
// sqr computed exactly as reference: |q|^2 + |p|^2 - 2 q.p
// ---------------------------------------------------------------------------
__global__ void k_ball_query(const float* __restrict__ xyz, const float* __restrict__ nxyz,
                             int B, int N, int S, int ns, float r2, int* __restrict__ idx)
{
  int i = blockIdx.x * blockDim.x + threadIdx.x;
  if (i >= B * S) return;
  int b = i / S;
  const float* q = nxyz + (size_t)i * 3;
  const float qx = q[0], qy = q[1], qz = q[2];
  const float sn = qx * qx + qy * qy + qz * qz;
  int* o = idx + (size_t)i * ns;
  int cnt = 0;
  const float* x = xyz + (size_t)b * N * 3;
  for (int n = 0; n < N && cnt < ns; ++n) {
    const float px = x[n * 3 + 0], py = x[n * 3 + 1], pz = x[n * 3 + 2];
    const float sx = px * px + py * py + pz * pz;
    const float dot = qx * px + qy * py + qz * pz;
    const float sqr = sn + sx - 2.f * dot;
    if (!(sqr > r2)) o[cnt++] = n;
  }
  const int first = (cnt > 0) ? o[0] : (N - 1);
  for (int j = cnt; j < ns; ++j) o[j] = first;
}

// ---------------------------------------------------------------------------
// Fused SA stage: one block per (b,s) group. Gather into LDS, run 3 WMMA GEMM
// layers (+bias+BN+ReLU) ping-ponging in LDS, max-pool over the neighborhood.
// Weights pre-padded; layer loop fully unrolled so weight pointers stay in the
// global address space. Dual-M tiles per wave share each B-fragment load.
// ---------------------------------------------------------------------------
__global__ void k_sa_fused(
    const float* __restrict__ xyz, const float* __restrict__ feats,
    const float* __restrict__ nxyz, const int* __restrict__ idx,
    int N, int S, int ns, int Cf, int cinPad, int bufCap,
    const float* __restrict__ w1, const float* __restrict__ b1,
    const float* __restrict__ g1, const float* __restrict__ e1, int C1,
    const float* __restrict__ w2, const float* __restrict__ b2,
    const float* __restrict__ g2, const float* __restrict__ e2, int C2,
    const float* __restrict__ w3, const float* __restrict__ b3,
    const float* __restrict__ g3, const float* __restrict__ e3, int C3,
    float* __restrict__ outf)
{
  extern __shared__ float smem[];
  float* bufA = smem;
  float* bufB = smem + bufCap;
  const int gs = blockIdx.x;   // b*S + s
  const int b = gs / S;
  const int tid = threadIdx.x;
  const int lane = tid & 31;
  const int wave = tid >> 5;
  const int nwaves = blockDim.x >> 5;
  const int* gi = idx + (size_t)gs * ns;
  const float* q = nxyz + (size_t)gs * 3;

  // gather: grouped = [xyz[k]-new_xyz (3) | feats[k] (Cf) | zero pad]
  for (int p = tid; p < ns * cinPad; p += blockDim.x) {
    const int j = p / cinPad;
    const int c = p - j * cinPad;
    const int k = gi[j];
    float v = 0.f;
    if (c < 3)             v = xyz[((size_t)b * N + k) * 3 + c] - q[c];
    else if (c < 3 + Cf)   v = feats[((size_t)b * N + k) * Cf + (c - 3)];
    bufA[p] = v;
  }
  __syncthreads();

  const float* Wl[3] = { w1, w2, w3 };
  const float* Bl[3] = { b1, b2, b3 };
  const float* Gl[3] = { g1, g2, g3 };
  const float* El[3] = { e1, e2, e3 };
  const int    Cl[3] = { C1, C2, C3 };
  const int    Kl[3] = { cinPad, C1, C2 };   // padded K per layer

  float* in = bufA;
  float* out = bufB;
  const int Mt2 = ns >> 5;                   // M tile-pairs (ns = 32 or 64)
#pragma unroll
  for (int L = 0; L < 3; ++L) {
    const int Cout = Cl[L];
    const int kPad = Kl[L];
    const int Nt = Cout >> 4;
    const int tiles = Mt2 * Nt;
    for (int t = wave; t < tiles; t += nwaves) {
      const int mt = (t % Mt2) * 2;
      const int nt = t / Mt2;
      v8f acc0, acc1;
      wmma_tile2_f32(in, kPad, mt * 16, Wl[L], kPad, nt * 16, lane, acc0, acc1);
      wmma_epilogue(acc0, out, Cout, mt * 16,      nt * 16, Bl[L], Gl[L], El[L], Cout, 1, lane);
      wmma_epilogue(acc1, out, Cout, mt * 16 + 16, nt * 16, Bl[L], Gl[L], El[L], Cout, 1, lane);
    }
    __syncthreads();
    float* tmp = in; in = out; out = tmp;
  }

  // max-pool over the ns neighborhood points
  const int Cfin = Cl[2];
  for (int c = tid; c < Cfin; c += blockDim.x) {
    float m = -3.4e38f;
    for (int j = 0; j < ns; ++j) m = fmaxf(m, in[j * Cfin + c]);
    outf[(size_t)gs * Cfin + c] = m;
  }
}

// ---------------------------------------------------------------------------
// X3[b*128+n, 0:260] = [l2_xyz(3) | l2_f(256) | pad(1)]
// ---------------------------------------------------------------------------
__global__ void k_build_x3(const float* __restrict__ l2xyz, const float* __restrict__ l2f,
                           int B, float* __restrict__ X)
{
  int i = blockIdx.x * blockDim.x + threadIdx.x;
  int total = B * 128 * 260;
  if (i >= total) return;
  int c = i % 260;
  int p = i / 260;
  float v = 0.f;
  if (c < 3)        v = l2xyz[(size_t)p * 3 + c];
  else if (c < 259) v = l2f[(size_t)p * 256 + (c - 3)];
  X[i] = v;
}

// ---------------------------------------------------------------------------
// Generic WMMA GEMM: Y[P,Cout] = act(X[P,ldx] * Wp[CoutPad,kPad]^T + bias ; BN)
// One 32x16 (dual-M) output tile per wave. P multiple of 32 everywhere.
// ---------------------------------------------------------------------------
__global__ void k_gemm(const float* __restrict__ X, int ldx, int P,
                       const float* __restrict__ Wp, int kPad,
                       const float* __restrict__ bias,
                       const float* __restrict__ gamma, const float* __restrict__ beta,
                       float* __restrict__ Y, int Cout, int relu)
{
  const int lane = threadIdx.x & 31;
  const int wave = threadIdx.x >> 5;
  const int Mt2 = P >> 5;
  const int Nt = (Cout + 15) >> 4;
  const int t = blockIdx.x * (blockDim.x >> 5) + wave;
  if (t >= Mt2 * Nt) return;
  const int mt = (t % Mt2) * 2;
  const int nt = t / Mt2;
  v8f acc0, acc1;
  wmma_tile2_f32(X, ldx, mt * 16, Wp, kPad, nt * 16, lane, acc0, acc1);
  wmma_epilogue(acc0, Y, Cout, mt * 16,      nt * 16, bias, gamma, beta, Cout, relu, lane);
  wmma_epilogue(acc1, Y, Cout, mt * 16 + 16, nt * 16, bias, gamma, beta, Cout, relu, lane);
}

// ---------------------------------------------------------------------------
// SA3 max over 128 points per batch
// ---------------------------------------------------------------------------
__global__ void k_maxpool(const float* __restrict__ H, int groups, int npts, int C,
                          float* __restrict__ out)
{
  int i = blockIdx.x * blockDim.x + threadIdx.x;
  if (i >= groups * C) return;
  int c = i % C;
  int g = i / C;
  const float* h = H + (size_t)g * npts * C + c;
  float m = -3.4e38f;
  for (int j = 0; j < npts; ++j) m = fmaxf(m, h[(size_t)j * C]);
  out[i] = m;
}

// ---------------------------------------------------------------------------
// log_softmax over 40 classes per row
// ---------------------------------------------------------------------------
__global__ void k_logsoftmax(const float* __restrict__ logits, int B, int C,
                             float* __restrict__ out)
{
  int b = blockIdx.x * blockDim.x + threadIdx.x;
  if (b >= B) return;
  const float* x = logits + (size_t)b * C;
  float mx = x[0];
  for (int c = 1; c < C; ++c) mx = fmaxf(mx, x[c]);
  float s = 0.f;
  for (int c = 0; c < C; ++c) s += expf(x[c] - mx);
  const float ls = logf(s) + mx;
  for (int c = 0; c < C; ++c) out[(size_t)b * C + c] = x[c] - ls;
}

// ---------------------------------------------------------------------------
extern "C" void kernel_launch(void* const* d_in, const int* in_sizes, int n_in,
                              void* d_out, int out_size, void* d_ws, size_t ws_size,
                              hipStream_t stream)
{
  (void)in_sizes; (void)n_in; (void)out_size; (void)ws_size;
  const int B = 32, N = 2048;

  const float* points = (const float*)d_in[0];
  // 9 MLP layers: sa1[0..2], sa2[3..5], sa3[6..8]; each {w,b,gamma,beta}
  const float *w[9], *bi[9], *ga[9], *be[9];
  int ii = 1;
  for (int l = 0; l < 9; ++l) {
    w[l]  = (const float*)d_in[ii++];
    bi[l] = (const float*)d_in[ii++];
    ga[l] = (const float*)d_in[ii++];
    be[l] = (const float*)d_in[ii++];
  }
  const float* fc1w = (const float*)d_in[ii++];
  const float* fc1b = (const float*)d_in[ii++];
  const float* fc2w = (const float*)d_in[ii++];
  const float* fc2b = (const float*)d_in[ii++];
  const float* fc3w = (const float*)d_in[ii++];
  const float* fc3b = (const float*)d_in[ii++];
  const float* bn1g = (const float*)d_in[ii++];
  const float* bn1e = (const float*)d_in[ii++];
  const float* bn2g = (const float*)d_in[ii++];
  const float* bn2e = (const float*)d_in[ii++];

  // workspace carve-up (256B aligned)
  char* wsp = (char*)d_ws;
  size_t off = 0;
  auto alloc = [&](size_t bytes) -> void* {
    void* p = wsp + off;
    off += (bytes + 255) & ~(size_t)255;
    return p;
  };
  float* xyz  = (float*)alloc((size_t)B * N * 3 * 4);
  float* nrm  = (float*)alloc((size_t)B * N * 3 * 4);
  int*   fps1 = (int*)  alloc((size_t)B * 512 * 4);
  float* nx1  = (float*)alloc((size_t)B * 512 * 3 * 4);
  int*   idx1 = (int*)  alloc((size_t)B * 512 * 32 * 4);
  float* l1f  = (float*)alloc((size_t)B * 512 * 128 * 4);
  int*   fps2 = (int*)  alloc((size_t)B * 128 * 4);
  float* nx2  = (float*)alloc((size_t)B * 128 * 3 * 4);
  int*   idx2 = (int*)  alloc((size_t)B * 128 * 64 * 4);
  float* l2f  = (float*)alloc((size_t)B * 128 * 256 * 4);
  float* X3   = (float*)alloc((size_t)B * 128 * 260 * 4);
  float* H3a  = (float*)alloc((size_t)B * 128 * 256 * 4);
  float* H3b  = (float*)alloc((size_t)B * 128 * 512 * 4);
  float* H3c  = (float*)alloc((size_t)B * 128 * 1024 * 4);
  float* l3f  = (float*)alloc((size_t)B * 1024 * 4);
  float* hx1  = (float*)alloc((size_t)B * 512 * 4);
  float* hx2  = (float*)alloc((size_t)B * 256 * 4);
  float* lg   = (float*)alloc((size_t)B * 40 * 4);

  // padded weights: {cin, Cout, kPad, CoutPad} per layer
  const int wcin[12]  = { 6, 64, 64, 131, 128, 128, 259, 256, 512, 1024, 512, 256 };
  const int wcout[12] = { 64, 64, 128, 128, 128, 256, 256, 512, 1024, 512, 256, 40 };
  const int wkp[12]   = { 8, 64, 64, 132, 128, 128, 260, 256, 512, 1024, 512, 256 };
  const int wnp[12]   = { 64, 64, 128, 128, 128, 256, 256, 512, 1024, 512, 256, 48 };
  const float* wsrc[12] = { w[0], w[1], w[2], w[3], w[4], w[5], w[6], w[7], w[8],
                            fc1w, fc2w, fc3w };
  float* wp[12];
  for (int l = 0; l < 12; ++l) {
    wp[l] = (float*)alloc((size_t)wnp[l] * wkp[l] * 4);
  }
  for (int l = 0; l < 12; ++l) {
    const int total = wnp[l] * wkp[l];
    k_pad_weights<<<(total + 255) / 256, 256, 0, stream>>>(
        wsrc[l], wcin[l], wcout[l], wp[l], wkp[l], wnp[l]);
  }

  auto gemm_grid = [](int P, int Cout) {
    const int t = (P >> 5) * ((Cout + 15) >> 4);   // dual-M tiles
    return (t + 7) / 8;                            // 8 waves per block
  };

  // --- split points into xyz / normals ---
  {
    const int total = B * N * 3;
    k_split_points<<<(total + 255) / 256, 256, 0, stream>>>(points, xyz, nrm, B, N);
  }

  // --- SA1: npoint=512, radius=0.2, nsample=32, 6->64->64->128 ---
  k_fps<<<B, 256, 0, stream>>>(xyz, N, 512, fps1);
  {
    const int total = B * 512 * 3;
    k_gather_xyz<<<(total + 255) / 256, 256, 0, stream>>>(xyz, fps1, N, 512, B, nx1);
  }
  {
    const int total = B * 512;
    k_ball_query<<<(total + 255) / 256, 256, 0, stream>>>(xyz, nx1, B, N, 512, 32, 0.04f, idx1);
  }
  {
    const int bufCap = 32 * 128;                 // ns * maxC
    const int shmem = 2 * bufCap * 4;            // 32 KB
    k_sa_fused<<<B * 512, 256, shmem, stream>>>(
        xyz, nrm, nx1, idx1, N, 512, 32, 3, 8, bufCap,
        wp[0], bi[0], ga[0], be[0], 64,
        wp[1], bi[1], ga[1], be[1], 64,
        wp[2], bi[2], ga[2], be[2], 128,
        l1f);
  }

  // --- SA2: npoint=128, radius=0.4, nsample=64, 131->128->128->256 ---
  k_fps<<<B, 256, 0, stream>>>(nx1, 512, 128, fps2);
  {
    const int total = B * 128 * 3;
    k_gather_xyz<<<(total + 255) / 256, 256, 0, stream>>>(nx1, fps2, 512, 128, B, nx2);
  }
  {
    const int total = B * 128;
    k_ball_query<<<(total + 255) / 256, 256, 0, stream>>>(nx1, nx2, B, 512, 128, 64, 0.16f, idx2);
  }
  {
    const int bufCap = 64 * 256;                 // ns * maxC
    const int shmem = 2 * bufCap * 4;            // 128 KB (of 320 KB/WGP LDS)
    k_sa_fused<<<B * 128, 256, shmem, stream>>>(
        nx1, l1f, nx2, idx2, 512, 128, 64, 128, 132, bufCap,
        wp[3], bi[3], ga[3], be[3], 128,
        wp[4], bi[4], ga[4], be[4], 128,
        wp[5], bi[5], ga[5], be[5], 256,
        l2f);
  }

  // --- SA3: global group of 128 points, 259->256->512->1024, max over 128 ---
  {
    const int total = B * 128 * 260;
    k_build_x3<<<(total + 255) / 256, 256, 0, stream>>>(nx2, l2f, B, X3);
  }
  const int P3 = B * 128;   // 4096
  k_gemm<<<gemm_grid(P3, 256), 256, 0, stream>>>(X3, 260, P3, wp[6], 260,
                                                 bi[6], ga[6], be[6], H3a, 256, 1);
  k_gemm<<<gemm_grid(P3, 512), 256, 0, stream>>>(H3a, 256, P3, wp[7], 256,
                                                 bi[7], ga[7], be[7], H3b, 512, 1);
  k_gemm<<<gemm_grid(P3, 1024), 256, 0, stream>>>(H3b, 512, P3, wp[8], 512,
                                                  bi[8], ga[8], be[8], H3c, 1024, 1);
  {
    const int total = B * 1024;
    k_maxpool<<<(total + 255) / 256, 256, 0, stream>>>(H3c, B, 128, 1024, l3f);
  }

  // --- Head: 1024->512->256->40, log_softmax ---
  k_gemm<<<gemm_grid(B, 512), 256, 0, stream>>>(l3f, 1024, B, wp[9], 1024,
                                                fc1b, bn1g, bn1e, hx1, 512, 1);
  k_gemm<<<gemm_grid(B, 256), 256, 0, stream>>>(hx1, 512, B, wp[10], 512,
                                                fc2b, bn2g, bn2e, hx2, 256, 1);
  k_gemm<<<gemm_grid(B, 40), 256, 0, stream>>>(hx2, 256, B, wp[11], 256,
                                               fc3b, nullptr, nullptr, lg, 40, 0);
  k_logsoftmax<<<1, 32, 0, stream>>>(lg, B, 40, (float*)d_out);
}